// GAT_31353261261182
// MI455X (gfx1250) — compile-verified
//
#include <hip/hip_runtime.h>
#include <hip/hip_bf16.h>

// ---------------- problem constants (match reference) ----------------
#define N_NODES  50000
#define N_EDGES  800000
#define C_IN     128
#define HIDF     128
#define C_OUTF   64
#define NGRAPH   64
#define NEG_SLOPE 0.2f
#define EPSV      1e-16f

typedef __attribute__((ext_vector_type(2))) float v2f;
typedef __attribute__((ext_vector_type(8))) float v8f;

// Monotone bijection float -> u32 so unsigned max == float max (handles negatives).
__device__ __forceinline__ unsigned ford(float f) {
    unsigned u = __float_as_uint(f);
    return (u & 0x80000000u) ? ~u : (u | 0x80000000u);
}
__device__ __forceinline__ float fordinv(unsigned u) {
    u = (u & 0x80000000u) ? (u ^ 0x80000000u) : ~u;
    return __uint_as_float(u);
}
__device__ __forceinline__ float lrelu(float x) { return x > 0.f ? x : NEG_SLOPE * x; }

// ---------------- fp32 WMMA GEMM: C[MxNCOLS] = A[MxK] @ B[KxNCOLS] ----------------
// 256-thread block = 8 waves; each wave owns one 16-row strip of C and keeps
// ALL NCOLS/16 accumulator tiles in registers, so A is streamed exactly once.
// B (<= 64 KB) is staged in LDS once per block and read via ds_load.
// v_wmma_f32_16x16x4_f32 gives exact fp32 numerics. Requires M % 16 == 0.
template <int K, int NCOLS>
__global__ __launch_bounds__(256) void gat_gemm_f32_wmma(
    const float* __restrict__ A, const float* __restrict__ B,
    float* __restrict__ C, int M) {
    constexpr int NT = NCOLS / 16;
    __shared__ float Bs[K * NCOLS];           // 64 KB max (K=128, NCOLS=128)

    for (int i = threadIdx.x; i < K * NCOLS; i += 256) Bs[i] = B[i];
    __syncthreads();

    const int wave = threadIdx.x >> 5;
    const int lane = threadIdx.x & 31;
    const int half = lane >> 4;               // 0: K-pair {0,1}, 1: K-pair {2,3}
    const int l16  = lane & 15;
    const int m0   = (blockIdx.x * 8 + wave) * 16;
    if (m0 >= M) return;                      // whole-wave uniform; EXEC stays full

    v8f acc[NT];
#pragma unroll
    for (int t = 0; t < NT; ++t) acc[t] = (v8f){};

    const float* arow = A + (size_t)(m0 + l16) * K + 2 * half;
    const float* brow = Bs + (size_t)(2 * half) * NCOLS + l16;

    for (int k = 0; k < K; k += 4) {
        v2f a;
        a.x = arow[k];
        a.y = arow[k + 1];
        const float* bk = brow + (size_t)k * NCOLS;
#pragma unroll
        for (int t = 0; t < NT; ++t) {
            v2f b;
            b.x = bk[t * 16];
            b.y = bk[t * 16 + NCOLS];
            acc[t] = __builtin_amdgcn_wmma_f32_16x16x4_f32(
                /*neg_a=*/false, a, /*neg_b=*/false, b,
                /*c_mod=*/(short)0, acc[t], /*reuse_a=*/false, /*reuse_b=*/false);
        }
    }

    // C/D layout: lane<16 -> M = v, lane>=16 -> M = v+8 ; N = l16 (+16*t)
    float* crow = C + (size_t)(m0 + 8 * half) * NCOLS + l16;
#pragma unroll
    for (int t = 0; t < NT; ++t)
#pragma unroll
        for (int v = 0; v < 8; ++v)
            crow[(size_t)v * NCOLS + t * 16] = acc[t][v];
}

// ---------------- per-node attention scores: s = h[i] . a ----------------
__global__ void gat_node_scores(const float* __restrict__ h,
                                const float* __restrict__ a_s,
                                const float* __restrict__ a_d,
                                float* __restrict__ ss, float* __restrict__ sd,
                                int n, int F) {
    const int wid  = blockIdx.x * (blockDim.x >> 5) + (threadIdx.x >> 5);
    const int lane = threadIdx.x & 31;
    if (wid >= n) return;
    float s0 = 0.f, s1 = 0.f;
    const float* row = h + (size_t)wid * F;
    for (int f = lane; f < F; f += 32) {
        const float v = row[f];
        s0 += v * a_s[f];
        s1 += v * a_d[f];
    }
#pragma unroll
    for (int off = 16; off; off >>= 1) {
        s0 += __shfl_xor(s0, off, 32);
        s1 += __shfl_xor(s1, off, 32);
    }
    if (lane == 0) { ss[wid] = s0; sd[wid] = s1; }
}

// ---------------- fills ----------------
__global__ void gat_fill_f32(float* __restrict__ p, float v, size_t n) {
    size_t i = (size_t)blockIdx.x * blockDim.x + threadIdx.x;
    if (i < n) p[i] = v;
}
__global__ void gat_fill_u32(unsigned* __restrict__ p, unsigned v, size_t n) {
    size_t i = (size_t)blockIdx.x * blockDim.x + threadIdx.x;
    if (i < n) p[i] = v;
}

// ---------------- edge passes (per-edge state carried in ealpha) ----------------
__device__ __forceinline__ void edge_sd(const int* __restrict__ ei, int e, int nE,
                                        int& s, int& d) {
    if (e < nE) { s = ei[e]; d = ei[nE + e]; }
    else        { s = e - nE; d = s; }       // appended self-loops
}

// pass 1: alpha = leaky_relu(ss[src] + sd[dst]); store it; segment max over dst
__global__ void gat_edge_max(const int* __restrict__ ei,
                             const float* __restrict__ ss, const float* __restrict__ sd,
                             float* __restrict__ ealpha, unsigned* __restrict__ amax,
                             int nE, int nT) {
    const int e = blockIdx.x * blockDim.x + threadIdx.x;
    if (e >= nT) return;
    int s, d; edge_sd(ei, e, nE, s, d);
    const float a = lrelu(ss[s] + sd[d]);
    ealpha[e] = a;
    atomicMax(&amax[d], ford(a));
}

// pass 2: ealpha <- exp(alpha - amax[dst]); segment sum over dst
__global__ void gat_edge_sum(const int* __restrict__ ei,
                             float* __restrict__ ealpha,
                             const unsigned* __restrict__ amax, float* __restrict__ denom,
                             int nE, int nT) {
    const int e = blockIdx.x * blockDim.x + threadIdx.x;
    if (e >= nT) return;
    const int d = (e < nE) ? ei[nE + e] : (e - nE);
    const float ex = expf(ealpha[e] - fordinv(amax[d]));
    ealpha[e] = ex;
    atomicAdd(&denom[d], ex);
}

// pass 3 (one wave per edge): att = e / (denom+eps); acc[dst] += att * h[src]
__global__ void gat_edge_scatter(const int* __restrict__ ei,
                                 const float* __restrict__ ealpha,
                                 const float* __restrict__ denom,
                                 const float* __restrict__ h, float* __restrict__ acc,
                                 int nE, int nT, int F) {
    const int wid  = blockIdx.x * (blockDim.x >> 5) + (threadIdx.x >> 5);
    const int lane = threadIdx.x & 31;
    if (wid >= nT) return;
    int s, d; edge_sd(ei, wid, nE, s, d);
    const float att = ealpha[wid] / (denom[d] + EPSV);
    const float* hs = h + (size_t)s * F;
    float* ad       = acc + (size_t)d * F;
    for (int f = lane; f < F; f += 32) atomicAdd(&ad[f], att * hs[f]);
}

// ---------------- bias (+ optional relu), in place ----------------
__global__ void gat_bias_act(float* __restrict__ x, const float* __restrict__ b,
                             size_t total, int Fmask, int do_relu) {
    size_t i = (size_t)blockIdx.x * blockDim.x + threadIdx.x;
    if (i >= total) return;
    float v = x[i] + b[i & (size_t)Fmask];
    x[i] = do_relu ? fmaxf(v, 0.f) : v;
}

// ---------------- global mean pool (accumulate) ----------------
__global__ void gat_pool(const float* __restrict__ h, const int* __restrict__ batch,
                         float* __restrict__ pooled, float* __restrict__ cnt, int n) {
    const int i = blockIdx.x;
    const int c = threadIdx.x;
    if (i >= n) return;
    const int b = batch[i];
    atomicAdd(&pooled[b * C_OUTF + c], h[(size_t)i * C_OUTF + c]);
    if (c == 0) atomicAdd(&cnt[b], 1.f);
}

// ---------------- mean + log_softmax over classes ----------------
__global__ void gat_logsoftmax(const float* __restrict__ pooled,
                               const float* __restrict__ cnt,
                               float* __restrict__ out) {
    __shared__ float sh[C_OUTF];
    const int b = blockIdx.x;
    const int c = threadIdx.x;
    const float v = pooled[b * C_OUTF + c] / fmaxf(cnt[b], 1.f);
    sh[c] = v;
    __syncthreads();
    for (int s = C_OUTF / 2; s > 0; s >>= 1) {
        if (c < s) sh[c] = fmaxf(sh[c], sh[c + s]);
        __syncthreads();
    }
    const float mx = sh[0];
    __syncthreads();
    sh[c] = expf(v - mx);
    __syncthreads();
    for (int s = C_OUTF / 2; s > 0; s >>= 1) {
        if (c < s) sh[c] += sh[c + s];
        __syncthreads();
    }
    out[b * C_OUTF + c] = v - mx - logf(sh[0]);
}

// ---------------- orchestration ----------------
extern "C" void kernel_launch(void* const* d_in, const int* in_sizes, int n_in,
                              void* d_out, int out_size, void* d_ws, size_t ws_size,
                              hipStream_t stream) {
    const float* x    = (const float*)d_in[0];
    const int*   ei   = (const int*)  d_in[1];   // [2, E] flat: row0=src, row1=dst
    const int*   bat  = (const int*)  d_in[2];
    const float* W1   = (const float*)d_in[3];
    const float* as1  = (const float*)d_in[4];
    const float* ad1  = (const float*)d_in[5];
    const float* b1   = (const float*)d_in[6];
    const float* W2   = (const float*)d_in[7];
    const float* as2  = (const float*)d_in[8];
    const float* ad2  = (const float*)d_in[9];
    const float* b2   = (const float*)d_in[10];
    float* out = (float*)d_out;

    const int N  = N_NODES;
    const int E  = N_EDGES;
    const int ET = E + N;               // with self-loops

    // workspace carve-up
    float*    h1     = (float*)d_ws;                 // N*128
    float*    acc1   = h1   + (size_t)N * HIDF;      // N*128
    float*    h2     = acc1 + (size_t)N * HIDF;      // N*64
    float*    acc2   = h2   + (size_t)N * C_OUTF;    // N*64
    float*    ealpha = acc2 + (size_t)N * C_OUTF;    // ET
    float*    ssrc   = ealpha + ET;                  // N
    float*    sdst   = ssrc + N;                     // N
    unsigned* amax   = (unsigned*)(sdst + N);        // N
    float*    denom  = (float*)(amax + N);           // N
    float*    pooled = denom + N;                    // B*64
    float*    cnt    = pooled + NGRAPH * C_OUTF;     // B

    const int TB = 256;
    const int edgeGrid  = (ET + TB - 1) / TB;
    const int waveGrid  = (ET + (TB / 32) - 1) / (TB / 32);   // wave-per-edge kernels
    const int nodeWaves = (N + (TB / 32) - 1) / (TB / 32);
    const int gemmGrid  = ((N / 16) + 7) / 8;                  // 8 row-strips per block

    // ===== Layer 1 =====
    gat_gemm_f32_wmma<C_IN, HIDF><<<gemmGrid, 256, 0, stream>>>(x, W1, h1, N);
    gat_node_scores<<<nodeWaves, TB, 0, stream>>>(h1, as1, ad1, ssrc, sdst, N, HIDF);

    {
        size_t n = (size_t)N * HIDF;
        gat_fill_f32<<<(int)((n + TB - 1) / TB), TB, 0, stream>>>(acc1, 0.f, n);
        gat_fill_u32<<<(N + TB - 1) / TB, TB, 0, stream>>>(amax, 0u, (size_t)N);
        gat_fill_f32<<<(N + TB - 1) / TB, TB, 0, stream>>>(denom, 0.f, (size_t)N);
    }
    gat_edge_max    <<<edgeGrid, TB, 0, stream>>>(ei, ssrc, sdst, ealpha, amax, E, ET);
    gat_edge_sum    <<<edgeGrid, TB, 0, stream>>>(ei, ealpha, amax, denom, E, ET);
    gat_edge_scatter<<<waveGrid, TB, 0, stream>>>(ei, ealpha, denom, h1, acc1, E, ET, HIDF);
    {
        size_t n = (size_t)N * HIDF;
        gat_bias_act<<<(int)((n + TB - 1) / TB), TB, 0, stream>>>(acc1, b1, n, HIDF - 1, 1);
    }

    // ===== Layer 2 =====
    gat_gemm_f32_wmma<HIDF, C_OUTF><<<gemmGrid, 256, 0, stream>>>(acc1, W2, h2, N);
    gat_node_scores<<<nodeWaves, TB, 0, stream>>>(h2, as2, ad2, ssrc, sdst, N, C_OUTF);

    {
        size_t n = (size_t)N * C_OUTF;
        gat_fill_f32<<<(int)((n + TB - 1) / TB), TB, 0, stream>>>(acc2, 0.f, n);
        gat_fill_u32<<<(N + TB - 1) / TB, TB, 0, stream>>>(amax, 0u, (size_t)N);
        gat_fill_f32<<<(N + TB - 1) / TB, TB, 0, stream>>>(denom, 0.f, (size_t)N);
    }
    gat_edge_max    <<<edgeGrid, TB, 0, stream>>>(ei, ssrc, sdst, ealpha, amax, E, ET);
    gat_edge_sum    <<<edgeGrid, TB, 0, stream>>>(ei, ealpha, amax, denom, E, ET);
    gat_edge_scatter<<<waveGrid, TB, 0, stream>>>(ei, ealpha, denom, h2, acc2, E, ET, C_OUTF);
    {
        size_t n = (size_t)N * C_OUTF;
        gat_bias_act<<<(int)((n + TB - 1) / TB), TB, 0, stream>>>(acc2, b2, n, C_OUTF - 1, 0);
    }

    // ===== Pool + log_softmax =====
    gat_fill_f32<<<(NGRAPH * C_OUTF + TB - 1) / TB, TB, 0, stream>>>(pooled, 0.f,
                                                                     (size_t)NGRAPH * C_OUTF);
    gat_fill_f32<<<1, TB, 0, stream>>>(cnt, 0.f, (size_t)NGRAPH);
    gat_pool<<<N, C_OUTF, 0, stream>>>(acc2, bat, pooled, cnt, N);
    gat_logsoftmax<<<NGRAPH, C_OUTF, 0, stream>>>(pooled, cnt, out);
}